// MultiCenterLoss_71476845740091
// MI455X (gfx1250) — compile-verified
//
#include <hip/hip_runtime.h>
#include <hip/hip_bf16.h>

typedef __attribute__((ext_vector_type(2))) float v2f;
typedef __attribute__((ext_vector_type(8))) float v8f;

#define B_ROWS   524288
#define D_DIM    64
#define NTILES   (B_ROWS / 16)        // 32768 tiles of 16 rows
#define NBLOCKS  1024
#define NTHREADS 256
#define WPB      (NTHREADS / 32)      // 8 waves per block

// Constant-lane broadcast via v_readlane_b32 (no LDS pipeline, no dscnt).
__device__ __forceinline__ float rdlane(float v, int srclane) {
    return __builtin_bit_cast(float,
        __builtin_amdgcn_readlane(__builtin_bit_cast(int, v), srclane));
}

// Main pass: one sweep over x. Per 16-row tile:
//   dot & rowsum via V_WMMA_F32_16X16X4_F32 chain (16 K-chunks),
//   xsq via per-lane FMA + shfl_xor(16),
//   readlane extraction, 3-center softmax tail on lanes 0..15.
__global__ __launch_bounds__(NTHREADS)
void mcl_main_kernel(const float* __restrict__ x,
                     const long long* __restrict__ labels,
                     const float* __restrict__ lin_w,
                     const float* __restrict__ lin_b,
                     const float* __restrict__ linear_p,
                     const float* __restrict__ bias_p,
                     const float* __restrict__ centers,
                     float* __restrict__ partial)
{
    __shared__ float s_w[64];
    __shared__ float s_red[WPB];

    const int tid  = threadIdx.x;
    const int lane = tid & 31;
    const int wid  = tid >> 5;

    if (tid < 64) s_w[tid] = lin_w[tid];
    __syncthreads();

    // Lane roles under the 32-bit 16x4 A/B fragment layout:
    //   row/col index n = lane%16, K sub-pair kb = 2*(lane/16).
    const int n  = lane & 15;
    const int kb = (lane >> 4) << 1;  // 0 or 2

    // B fragments are tile-invariant: col0 = w chunk, col1 = ones, rest 0.
    float b0[16], b1[16];
#pragma unroll
    for (int kc = 0; kc < 16; ++kc) {
        float w0 = s_w[4 * kc + kb];
        float w1 = s_w[4 * kc + kb + 1];
        b0[kc] = (n == 0) ? w0 : ((n == 1) ? 1.0f : 0.0f);
        b1[kc] = (n == 0) ? w1 : ((n == 1) ? 1.0f : 0.0f);
    }
    const float linb = lin_b[0];

    const int gwave  = blockIdx.x * WPB + wid;
    const int nwaves = NBLOCKS * WPB;

    float acc = 0.0f;

    for (int tile = gwave; tile < NTILES; tile += nwaves) {
        const int row_base = tile << 4;
        const float* xrow = x + (size_t)(row_base + n) * D_DIM + kb;

        v8f  c1 = {};     // col0 accumulates dot(x,w), col1 accumulates sum(x)
        float sq = 0.0f;  // per-lane half-row sum of squares
#pragma unroll
        for (int kc = 0; kc < 16; ++kc) {
            v2f a = *(const v2f*)(xrow + 4 * kc);     // global_load_b64, fragment layout
            v2f b; b.x = b0[kc]; b.y = b1[kc];
            c1 = __builtin_amdgcn_wmma_f32_16x16x4_f32(
                     false, a, false, b, (short)0, c1, false, false);
            sq = fmaf(a.y, a.y, fmaf(a.x, a.x, sq));
        }
        sq += __shfl_xor(sq, 16);  // combine half-rows -> full xsq, valid in lanes 0..15

        // D layout: VGPR v, lanes 0-15 -> (M=v, N=lane); lanes 16-31 -> (M=v+8, N=lane-16).
        // Route (dot, xs) of row r into lane r via v_readlane + cndmask:
        // branch-free, no LDS pipeline, waves stay fully independent.
        float dot = 0.0f, xs = 0.0f;
#pragma unroll
        for (int v = 0; v < 8; ++v) {
            const float dl = rdlane(c1[v], 0);    // D[v][0]
            const float sl = rdlane(c1[v], 1);    // D[v][1]
            const float dh = rdlane(c1[v], 16);   // D[v+8][0]
            const float sh = rdlane(c1[v], 17);   // D[v+8][1]
            const bool lo = (lane == v);
            const bool hi = (lane == v + 8);
            dot = lo ? dl : (hi ? dh : dot);
            xs  = lo ? sl : (hi ? sh : xs);
        }

        if (lane < 16) {
            const int   row = row_base + lane;
            const float xsq = sq;

            const int li = 3 * (int)labels[row];
            const float maps = dot + linb;

            const float lp0 = linear_p[li + 0], lp1 = linear_p[li + 1], lp2 = linear_p[li + 2];
            const float bp0 = bias_p[li + 0],   bp1 = bias_p[li + 1],   bp2 = bias_p[li + 2];
            const float cc0 = centers[li + 0],  cc1 = centers[li + 1],  cc2 = centers[li + 2];

            const float lg0 = fmaf(maps, lp0, bp0);
            const float lg1 = fmaf(maps, lp1, bp1);
            const float lg2 = fmaf(maps, lp2, bp2);
            const float m   = fmaxf(lg0, fmaxf(lg1, lg2));
            const float e0  = expf(lg0 - m);
            const float e1  = expf(lg1 - m);
            const float e2  = expf(lg2 - m);
            const float inv = 1.0f / (e0 + e1 + e2);

            const float sd0 = xsq - 2.0f * cc0 * xs + 64.0f * cc0 * cc0;
            const float sd1 = xsq - 2.0f * cc1 * xs + 64.0f * cc1 * cc1;
            const float sd2 = xsq - 2.0f * cc2 * xs + 64.0f * cc2 * cc2;

            acc += inv * (e0 * sd0 + e1 * sd1 + e2 * sd2);
        }
    }

    // Deterministic block reduction -> partial[blockIdx.x]
#pragma unroll
    for (int off = 16; off >= 1; off >>= 1) acc += __shfl_xor(acc, off);
    if (lane == 0) s_red[wid] = acc;
    __syncthreads();
    if (tid == 0) {
        float t = 0.0f;
#pragma unroll
        for (int i = 0; i < WPB; ++i) t += s_red[i];
        partial[blockIdx.x] = t;
    }
}

// Fixed-order final reduction: fully deterministic, overwrites d_out.
__global__ __launch_bounds__(256)
void mcl_reduce_kernel(const float* __restrict__ partial, float* __restrict__ out)
{
    __shared__ float s[256];
    float v = 0.0f;
    for (int i = threadIdx.x; i < NBLOCKS; i += 256) v += partial[i];
    s[threadIdx.x] = v;
    __syncthreads();
    for (int off = 128; off > 0; off >>= 1) {
        if (threadIdx.x < off) s[threadIdx.x] += s[threadIdx.x + off];
        __syncthreads();
    }
    if (threadIdx.x == 0) out[0] = s[0] * (1.0f / (float)B_ROWS);
}

extern "C" void kernel_launch(void* const* d_in, const int* in_sizes, int n_in,
                              void* d_out, int out_size, void* d_ws, size_t ws_size,
                              hipStream_t stream)
{
    const float*     x        = (const float*)d_in[0];
    const long long* labels   = (const long long*)d_in[1];   // int64 labels
    const float*     lin_w    = (const float*)d_in[2];
    const float*     lin_b    = (const float*)d_in[3];
    const float*     linear_p = (const float*)d_in[4];
    const float*     bias_p   = (const float*)d_in[5];
    const float*     centers  = (const float*)d_in[6];
    float*           out      = (float*)d_out;
    float*           partial  = (float*)d_ws;                // NBLOCKS floats (4 KB)

    mcl_main_kernel<<<NBLOCKS, NTHREADS, 0, stream>>>(
        x, labels, lin_w, lin_b, linear_p, bias_p, centers, partial);
    mcl_reduce_kernel<<<1, 256, 0, stream>>>(partial, out);
}